// ReLULocalZeroToken_82197084111407
// MI455X (gfx1250) — compile-verified
//
#include <hip/hip_runtime.h>
#include <stdint.h>

typedef float v2f __attribute__((ext_vector_type(2)));
typedef float v8f __attribute__((ext_vector_type(8)));
typedef int   v4i __attribute__((ext_vector_type(4)));

#define T_TOKENS 16384   // B*S
#define H_DIM    1024
#define DFF_DIM  4096

// ---- CDNA5 async global->LDS path (guarded; falls back to sync staging) ----
#if defined(__has_builtin)
#  if __has_builtin(__builtin_amdgcn_global_load_async_to_lds_b128)
#    define HAVE_ASYNC 1
#  endif
#endif
#ifndef HAVE_ASYNC
#  define HAVE_ASYNC 0
#endif

#if defined(__has_builtin) && __has_builtin(__builtin_amdgcn_s_wait_asynccnt)
#  define WAIT_ASYNC(n) __builtin_amdgcn_s_wait_asynccnt(n)
#else
#  define WAIT_ASYNC(n) asm volatile("s_wait_asynccnt %0" :: "n"(n) : "memory")
#endif

// builtin signature (from hipcc diagnostic): param0 = v4i* (generic), param1 = v4i addrspace(3)*
#define GPTR(p) ((v4i*)(p))
#define LPTR(p) ((__attribute__((address_space(3))) v4i*)(p))

// ---------------- gate: w = sigmoid(h@Wg + bg), mask = (logit>=0) | (label==-100) ------------
__global__ __launch_bounds__(256) void gate_kernel(
    const float* __restrict__ h, const float* __restrict__ Wg,
    const float* __restrict__ bg, const long long* __restrict__ labels,
    float* __restrict__ w_out, int* __restrict__ mask_out)
{
    const int wid  = threadIdx.x >> 5;
    const int lane = threadIdx.x & 31;
    const int t    = blockIdx.x * 8 + wid;      // one wave32 per token

    const float* hrow = h + (size_t)t * H_DIM;
    float acc = 0.f;
#pragma unroll
    for (int i = 0; i < H_DIM / 32; ++i) {
        const int idx = lane + i * 32;          // coalesced across lanes
        acc = fmaf(hrow[idx], Wg[idx], acc);
    }
#pragma unroll
    for (int off = 16; off > 0; off >>= 1)
        acc += __shfl_xor(acc, off, 32);

    const float logit = acc + bg[0];
    if (lane == 0) {
        w_out[t]    = 1.0f / (1.0f + __expf(-logit));
        mask_out[t] = ((logit >= 0.0f) || (labels[t] == -100ll)) ? 1 : 0;
    }
}

// ---------------- fp32 WMMA GEMM: C[T,N] = epilogue(A[T,K] @ B[K,N]) ------------------------
// Block: 256 threads = 8 waves (2x4). Block tile 128x128, K slab 16 (4 WMMA k-steps).
// Wave tile 64x32 = 4x2 accumulators of v_wmma_f32_16x16x4_f32.
// Double-buffered LDS fed by global_load_async_to_lds_b128 (ASYNCcnt pipeline).
template <int KDIM, int NDIM, bool SCALE_EPI, bool MASK_OUT, bool RELU>
__global__ __launch_bounds__(256) void wmma_gemm_kernel(
    const float* __restrict__ A, const float* __restrict__ Bm,
    const float* __restrict__ bias, const float* __restrict__ scl,
    const int* __restrict__ mask, float* __restrict__ C)
{
    constexpr int BM = 128, BN = 128, BK = 16;
    constexpr int ASTR = BK + 4;   // 20 floats: even stride -> 8B-aligned v2f reads, no 16-lane conflicts
    constexpr int BSTR = BN + 4;   // 132 floats

    __shared__ __align__(16) float As[2][BM][ASTR];
    __shared__ __align__(16) float Bs[2][BK][BSTR];

    const int tid   = threadIdx.x;
    const int lane  = tid & 31;
    const int wid   = tid >> 5;
    const int wm    = wid >> 2;             // 0..1  -> 64-row slab
    const int wn    = wid & 3;              // 0..3  -> 32-col slab
    const int bm    = blockIdx.y;
    const int bn    = blockIdx.x;
    const int lhalf = (lane < 16) ? 0 : 2;  // K=0,1 vs K=2,3 halves (A and B layouts)
    const int lcol  = lane & 15;

    v8f acc[4][2];
#pragma unroll
    for (int mi = 0; mi < 4; ++mi)
#pragma unroll
        for (int ni = 0; ni < 2; ++ni)
            acc[mi][ni] = v8f{};

    // per-thread staging coordinates (16B chunks; 2 chunks each for A and B)
    const int ar0 = tid >> 2, ac0 = (tid & 3) << 2;              // A chunk 0
    const int ar1 = (tid + 256) >> 2, ac1 = ((tid + 256) & 3) << 2;
    const int br0 = tid >> 5, bc0 = (tid & 31) << 2;             // B chunk 0
    const int br1 = (tid + 256) >> 5, bc1 = ((tid + 256) & 31) << 2;

    auto compute_slab = [&](int bi) {
#pragma unroll
        for (int ks = 0; ks < 4; ++ks) {
            const int kb = ks * 4 + lhalf;
            v2f afrag[4];
            v2f bfrag[2];
#pragma unroll
            for (int mi = 0; mi < 4; ++mi) {
                const int r = wm * 64 + mi * 16 + lcol;
                afrag[mi] = *reinterpret_cast<const v2f*>(&As[bi][r][kb]);
            }
#pragma unroll
            for (int ni = 0; ni < 2; ++ni) {
                const int c = wn * 32 + ni * 16 + lcol;
                bfrag[ni][0] = Bs[bi][kb + 0][c];
                bfrag[ni][1] = Bs[bi][kb + 1][c];
            }
#pragma unroll
            for (int mi = 0; mi < 4; ++mi)
#pragma unroll
                for (int ni = 0; ni < 2; ++ni)
                    acc[mi][ni] = __builtin_amdgcn_wmma_f32_16x16x4_f32(
                        false, afrag[mi], false, bfrag[ni],
                        (short)0, acc[mi][ni], false, false);
        }
    };

#if HAVE_ASYNC
    auto issue_slab = [&](int k0, int bi) {
        __builtin_amdgcn_global_load_async_to_lds_b128(
            GPTR(A + (size_t)(bm * BM + ar0) * KDIM + k0 + ac0),
            LPTR(&As[bi][ar0][ac0]), 0, 0);
        __builtin_amdgcn_global_load_async_to_lds_b128(
            GPTR(A + (size_t)(bm * BM + ar1) * KDIM + k0 + ac1),
            LPTR(&As[bi][ar1][ac1]), 0, 0);
        __builtin_amdgcn_global_load_async_to_lds_b128(
            GPTR(Bm + (size_t)(k0 + br0) * NDIM + bn * BN + bc0),
            LPTR(&Bs[bi][br0][bc0]), 0, 0);
        __builtin_amdgcn_global_load_async_to_lds_b128(
            GPTR(Bm + (size_t)(k0 + br1) * NDIM + bn * BN + bc1),
            LPTR(&Bs[bi][br1][bc1]), 0, 0);
    };

    constexpr int NK = KDIM / BK;
    issue_slab(0, 0);
    for (int kk = 0; kk < NK; ++kk) {
        const int cur = kk & 1;
        if (kk + 1 < NK) {
            issue_slab((kk + 1) * BK, cur ^ 1);   // prefetch next slab into other buffer
            WAIT_ASYNC(4);                        // this wave's slab-kk copies have landed
        } else {
            WAIT_ASYNC(0);
        }
        __syncthreads();                          // all waves' copies visible
        compute_slab(cur);
        __syncthreads();                          // done reading before buffer reuse
    }
#else
    // synchronous fallback: single-buffered load/store staging
    for (int k0 = 0; k0 < KDIM; k0 += BK) {
        {
            const float4 a0 = *reinterpret_cast<const float4*>(
                A + (size_t)(bm * BM + ar0) * KDIM + k0 + ac0);
            const float4 a1 = *reinterpret_cast<const float4*>(
                A + (size_t)(bm * BM + ar1) * KDIM + k0 + ac1);
            *reinterpret_cast<float4*>(&As[0][ar0][ac0]) = a0;
            *reinterpret_cast<float4*>(&As[0][ar1][ac1]) = a1;
            const float4 b0 = *reinterpret_cast<const float4*>(
                Bm + (size_t)(k0 + br0) * NDIM + bn * BN + bc0);
            const float4 b1v = *reinterpret_cast<const float4*>(
                Bm + (size_t)(k0 + br1) * NDIM + bn * BN + bc1);
            *reinterpret_cast<float4*>(&Bs[0][br0][bc0]) = b0;
            *reinterpret_cast<float4*>(&Bs[0][br1][bc1]) = b1v;
        }
        __syncthreads();
        compute_slab(0);
        __syncthreads();
    }
#endif

    // ---- epilogue: gate-scale fold (GEMM1) / mask-zero (GEMM2), bias, store ----
#pragma unroll
    for (int mi = 0; mi < 4; ++mi) {
#pragma unroll
        for (int ni = 0; ni < 2; ++ni) {
            const int cg = bn * BN + wn * 32 + ni * 16 + lcol;
            const float bb = bias[cg];
#pragma unroll
            for (int i = 0; i < 8; ++i) {
                const int rg = bm * BM + wm * 64 + mi * 16 + i + ((lane < 16) ? 0 : 8);
                float v;
                if constexpr (SCALE_EPI) v = fmaf(scl[rg], acc[mi][ni][i], bb);
                else                     v = acc[mi][ni][i] + bb;
                if constexpr (RELU)      v = fmaxf(v, 0.0f);
                if constexpr (MASK_OUT)  v = mask[rg] ? v : 0.0f;
                C[(size_t)rg * NDIM + cg] = v;
            }
        }
    }
}

extern "C" void kernel_launch(void* const* d_in, const int* in_sizes, int n_in,
                              void* d_out, int out_size, void* d_ws, size_t ws_size,
                              hipStream_t stream) {
    (void)in_sizes; (void)n_in; (void)out_size; (void)ws_size;
    const float*     h      = (const float*)d_in[0];
    // d_in[1]=cos, d_in[2]=sin, d_in[10]=cu_seq_lens_q : unused by reference
    const float*     Wg     = (const float*)d_in[3];
    const float*     bg     = (const float*)d_in[4];
    const float*     W1     = (const float*)d_in[5];
    const float*     b1     = (const float*)d_in[6];
    const float*     W2     = (const float*)d_in[7];
    const float*     b2     = (const float*)d_in[8];
    const long long* labels = (const long long*)d_in[9];
    float*           out    = (float*)d_out;

    char*  ws       = (char*)d_ws;
    float* w_buf    = (float*)(ws);                       // T floats
    int*   mask_buf = (int*)  (ws + 64 * 1024);           // T ints
    float* act      = (float*)(ws + 128 * 1024);          // T*DFF floats (256 MB)

    gate_kernel<<<T_TOKENS / 8, 256, 0, stream>>>(h, Wg, bg, labels, w_buf, mask_buf);

    // act = relu(w ⊙ (h @ W1) + b1)   (gate scale folded into epilogue)
    wmma_gemm_kernel<H_DIM, DFF_DIM, true, false, true>
        <<<dim3(DFF_DIM / 128, T_TOKENS / 128), 256, 0, stream>>>(
            h, W1, b1, w_buf, nullptr, act);

    // out = mask ? (act @ W2 + b2) : 0
    wmma_gemm_kernel<DFF_DIM, H_DIM, false, true, false>
        <<<dim3(H_DIM / 128, T_TOKENS / 128), 256, 0, stream>>>(
            act, W2, b2, nullptr, mask_buf, out);
}